// ElemExtractor_23313082483343
// MI455X (gfx1250) — compile-verified
//
#include <hip/hip_runtime.h>
#include <hip/hip_bf16.h>

typedef __attribute__((ext_vector_type(2))) float v2f;
typedef __attribute__((ext_vector_type(8))) float v8f;

#define BQ 16
#define TQ 4096
#define DQ 768
#define SMAXQ 128
#define H1 384
#define H2 128
#define TCHUNK 256          // tokens per block (segsum)
#define DTHREADS 192        // 192 threads x 4 dims = 768 dims

// ---------------------------------------------------------------------------
// Kernel 0: zero the sent_em scratch buffer [B*SMAX, D] floats
// ---------------------------------------------------------------------------
__global__ void zero_kernel(float* __restrict__ p, int n) {
    int i = blockIdx.x * blockDim.x + threadIdx.x;
    if (i < n) p[i] = 0.0f;
}

// ---------------------------------------------------------------------------
// Kernel 1: segment sum. seg_ids are sorted per sample, so each segment is a
// contiguous token run. Each thread owns FOUR consecutive feature dims
// (one global_load_b128 per token -> 512B per wave per row) and walks a token
// chunk, accumulating the current run in registers and flushing with
// atomicAdd only at run boundaries (and chunk edges).
// grid = (B, T/TCHUNK), block = DTHREADS (6 waves)
// ---------------------------------------------------------------------------
__global__ void segsum_kernel(const float* __restrict__ hidden,
                              const int* __restrict__ seg_ids,
                              float* __restrict__ sent) {
    const int b  = blockIdx.x;
    const int tc = blockIdx.y;
    const int d4 = threadIdx.x;              // float4 index within a row

    const float4* h4 = (const float4*)(hidden + (size_t)b * TQ * DQ);
    const int*    s  = seg_ids + (size_t)b * TQ;
    float*        o  = sent   + (size_t)b * SMAXQ * DQ;

    const int t0 = tc * TCHUNK;
    const int t1 = t0 + TCHUNK;

    int    cur = s[t0];
    float4 acc = make_float4(0.f, 0.f, 0.f, 0.f);

    for (int t = t0; t < t1; ++t) {
        const int sg = s[t];                 // wave-uniform
        if (sg != cur) {                     // wave-uniform branch, rare
            float* dst = &o[(size_t)cur * DQ + d4 * 4];
            atomicAdd(dst + 0, acc.x);
            atomicAdd(dst + 1, acc.y);
            atomicAdd(dst + 2, acc.z);
            atomicAdd(dst + 3, acc.w);
            acc = make_float4(0.f, 0.f, 0.f, 0.f);
            cur = sg;
        }
        const float4 v = h4[(size_t)t * (DQ / 4) + d4];
        acc.x += v.x; acc.y += v.y; acc.z += v.z; acc.w += v.w;
    }
    {
        float* dst = &o[(size_t)cur * DQ + d4 * 4];
        atomicAdd(dst + 0, acc.x);
        atomicAdd(dst + 1, acc.y);
        atomicAdd(dst + 2, acc.z);
        atomicAdd(dst + 3, acc.w);
    }
}

// ---------------------------------------------------------------------------
// Kernel 2: fused MLP head over sent_em [2048, 768]:
//   x1 = relu(A @ W1 + b1)   [16, 384] per block   (24 waves, one N-tile each)
//   x2 = relu(x1 @ W2 + b2)  [16, 128]             ( 8 waves)
//   out = x2 @ W3 + b3       [16, 2]               (scalar, wave 0)
// grid = 128 blocks (one 16-row M tile each), block = 768 threads = 24 waves.
// Uses V_WMMA_F32_16X16X4_F32 (full fp32 precision).
//
// Fragment layouts (ISA 7.12.2, wave32):
//  A 16x4:  lane l (m=l&15, half=l>>4): v.x=A[m][k+2*half], v.y=A[m][k+2*half+1]
//  B 4x16:  lane l (n=l&15, half=l>>4): v.x=B[k+2*half][n], v.y=B[k+2*half+1][n]
//  C/D 16x16: c[r] at lane l -> row r+8*half, col l&15
// ---------------------------------------------------------------------------
__global__ void __launch_bounds__(768)
mlp_kernel(const float* __restrict__ sent,
           const float* __restrict__ W1, const float* __restrict__ b1,
           const float* __restrict__ W2, const float* __restrict__ b2,
           const float* __restrict__ W3, const float* __restrict__ b3,
           float* __restrict__ out) {
    __shared__ float As[16 * DQ];   // 48 KB: input tile
    __shared__ float X1[16 * H1];   // 24 KB
    __shared__ float X2[16 * H2];   //  8 KB

    const int tile = blockIdx.x;            // M tile: rows [16*tile, 16*tile+16)
    const int tid  = threadIdx.x;
    const int wave = tid >> 5;              // 0..23
    const int lane = tid & 31;
    const int m    = lane & 15;
    const int half = lane >> 4;             // 0: K+0/K+1 rows, 1: K+2/K+3 rows

    // Cooperative load of the 16x768 input tile (16 floats per thread, coalesced)
    {
        const float* src = sent + (size_t)tile * 16 * DQ;
        for (int i = tid; i < 16 * DQ; i += 768) As[i] = src[i];
    }
    __syncthreads();

    // ---------------- Layer 1: [16x768] @ [768x384] ----------------
    {
        const int n0 = wave * 16;           // all 24 waves active
        v8f c = {};
        for (int k = 0; k < DQ; k += 4) {
            const int ka = k + 2 * half;
            v2f a, bb;
            a.x  = As[m * DQ + ka];
            a.y  = As[m * DQ + ka + 1];
            bb.x = W1[(size_t)ka * H1 + n0 + m];
            bb.y = W1[(size_t)(ka + 1) * H1 + n0 + m];
            c = __builtin_amdgcn_wmma_f32_16x16x4_f32(
                    false, a, false, bb, (short)0, c, false, false);
        }
        const float bias = b1[n0 + m];
        #pragma unroll
        for (int r = 0; r < 8; ++r) {
            const int row = r + 8 * half;
            float v = c[r] + bias;
            X1[row * H1 + n0 + m] = v > 0.0f ? v : 0.0f;
        }
    }
    __syncthreads();

    // ---------------- Layer 2: [16x384] @ [384x128] ----------------
    if (wave < 8) {                          // wave-uniform: EXEC all-1s inside
        const int n0 = wave * 16;
        v8f c = {};
        for (int k = 0; k < H1; k += 4) {
            const int ka = k + 2 * half;
            v2f a, bb;
            a.x  = X1[m * H1 + ka];
            a.y  = X1[m * H1 + ka + 1];
            bb.x = W2[(size_t)ka * H2 + n0 + m];
            bb.y = W2[(size_t)(ka + 1) * H2 + n0 + m];
            c = __builtin_amdgcn_wmma_f32_16x16x4_f32(
                    false, a, false, bb, (short)0, c, false, false);
        }
        const float bias = b2[n0 + m];
        #pragma unroll
        for (int r = 0; r < 8; ++r) {
            const int row = r + 8 * half;
            float v = c[r] + bias;
            X2[row * H2 + n0 + m] = v > 0.0f ? v : 0.0f;
        }
    }
    __syncthreads();

    // ---------------- Layer 3: [16x128] @ [128x2], scalar ----------------
    if (tid < 32) {
        const int row = tid >> 1;
        const int col = tid & 1;
        float acc = b3[col];
        for (int k = 0; k < H2; ++k) {
            acc = fmaf(X2[row * H2 + k], W3[k * 2 + col], acc);
        }
        out[((size_t)tile * 16 + row) * 2 + col] = acc;
    }
}

// ---------------------------------------------------------------------------
extern "C" void kernel_launch(void* const* d_in, const int* in_sizes, int n_in,
                              void* d_out, int out_size, void* d_ws, size_t ws_size,
                              hipStream_t stream) {
    const float* hidden  = (const float*)d_in[0];
    const int*   seg_ids = (const int*)  d_in[1];
    const float* W1 = (const float*)d_in[2];
    const float* b1 = (const float*)d_in[3];
    const float* W2 = (const float*)d_in[4];
    const float* b2 = (const float*)d_in[5];
    const float* W3 = (const float*)d_in[6];
    const float* b3 = (const float*)d_in[7];
    float* out = (float*)d_out;

    float* sent = (float*)d_ws;               // [B*SMAX, D] = 6 MB scratch
    const int n_sent = BQ * SMAXQ * DQ;

    zero_kernel<<<(n_sent + 1023) / 1024, 1024, 0, stream>>>(sent, n_sent);

    dim3 gseg(BQ, TQ / TCHUNK);
    segsum_kernel<<<gseg, DTHREADS, 0, stream>>>(hidden, seg_ids, sent);

    mlp_kernel<<<(BQ * SMAXQ) / 16, 768, 0, stream>>>(
        sent, W1, b1, W2, b2, W3, b3, out);
}